// RadarEncV3_35450660061768
// MI455X (gfx1250) — compile-verified
//
#include <hip/hip_runtime.h>
#include <hip/hip_bf16.h>

typedef float v2f __attribute__((ext_vector_type(2)));
typedef float v8f __attribute__((ext_vector_type(8)));

#define B_DIM   2
#define D_DIM   64
#define R_DIM   256
#define E_DIM   37
#define A_DIM   107
#define POS     (E_DIM * A_DIM)     // 3959 flat (elev, azim) positions
#define ROWS    (B_DIM * R_DIM)     // 512 (batch, range) rows
#define KTOP    100
#define STRIDE_D (R_DIM * POS)      // element stride between doppler slices: 1013504
#define TILES_PER_ROW ((POS + 15) / 16)   // 248
#define POS_CHUNKS (POS / 4)        // 989 -> we async-copy 988 full 16B chunks
#define ASYNC_CHUNKS 988            // 988 * 4 = 3952 floats
#define ASYNC_REM (POS - ASYNC_CHUNKS * 4)  // 7 remainder floats

// ---------------------------------------------------------------------------
// Kernel 1: doppler mean via V_WMMA_F32_16X16X4_F32.
//   cube[row, pos] = (1/64) * sum_d radar[b, d, r, pos]
// One wave handles a 16-position tile of one row. A-matrix is the constant
// 1/64 (folds the divide into the MAC); B-matrix carries 4 doppler values
// x 16 positions per step; 16 accumulation steps cover D=64. All D rows are
// identical, so acc[0] of lanes 0..15 = the 16 means.
// ---------------------------------------------------------------------------
__global__ __launch_bounds__(256)
void doppler_mean_wmma(const float* __restrict__ radar, float* __restrict__ cube) {
  const int lane = threadIdx.x & 31;
  const int wave = threadIdx.x >> 5;
  const int waveGlobal = blockIdx.x * 8 + wave;          // 8 waves per block

  const int row  = waveGlobal / TILES_PER_ROW;           // wave-uniform
  const int tile = waveGlobal % TILES_PER_ROW;
  if (row >= ROWS) return;                               // whole wave exits -> EXEC stays all-1s

  const int b = row / R_DIM;
  const int r = row % R_DIM;
  const long long base = (long long)b * D_DIM * STRIDE_D + (long long)r * POS;

  const int pos  = tile * 16 + (lane & 15);
  const int posc = (pos < POS) ? pos : (POS - 1);        // clamp tail-tile loads (store masked)
  const int kHalf = lane >> 4;                           // 0: lanes 0-15, 1: lanes 16-31

  // A = 16x4 matrix of 1/64. Layout: lanes 0-15 hold K=0,1; lanes 16-31 K=2,3.
  v2f amat;
  amat[0] = 1.0f / 64.0f;
  amat[1] = 1.0f / 64.0f;

  v8f acc = {};
  const float* colBase = radar + base + posc;            // X[d=0, pos]

  for (int kb = 0; kb < 16; ++kb) {
    const int d0 = kb * 4;
    // B = 4x16 tile: VGPR j of lane L holds B[K = 2j + (L>>4), N = L&15]
    //             = X[d = d0 + 2j + kHalf, pos]
    const float* p0 = colBase + (long long)(d0 + kHalf)     * STRIDE_D;
    const float* p1 = colBase + (long long)(d0 + 2 + kHalf) * STRIDE_D;
    v2f bmat;
    bmat[0] = *p0;
    bmat[1] = *p1;

    if (kb < 15) {
      // prefetch next K-block (-> global_prefetch_b8)
      __builtin_prefetch(colBase + (long long)(d0 + 4 + kHalf) * STRIDE_D, 0, 1);
    }

    // D = A(1/64) x B + C  -> accumulates the doppler mean in the matrix pipe
    acc = __builtin_amdgcn_wmma_f32_16x16x4_f32(
        /*neg_a=*/false, amat, /*neg_b=*/false, bmat,
        /*c_mod=*/(short)0, acc, /*reuse_a=*/false, /*reuse_b=*/false);
  }

  // acc[0] = D[M = 0 or 8, N = lane&15]; all M identical -> mean[pos].
  if (lane < 16 && pos < POS) {
    cube[(long long)row * POS + pos] = acc[0];
  }
}

// ---------------------------------------------------------------------------
// Kernel 2: per-row top-100 with argmax extraction (sorted descending, ties
// broken by lower flat index, matching jax.lax.top_k). One block per row.
// Row staging uses the CDNA5 async global->LDS path (b128, ASYNCcnt).
// ---------------------------------------------------------------------------
__global__ __launch_bounds__(256)
void topk_rows(const float* __restrict__ cube,
               float* __restrict__ features, int* __restrict__ coords) {
  __shared__ float svals[POS];       // 15836 B staged row
  __shared__ float rval[256];
  __shared__ int   ridx[256];

  const int row = blockIdx.x;
  const int tid = threadIdx.x;
  const int b = row / R_DIM;
  const int r = row % R_DIM;

  const float* src = cube + (long long)row * POS;

  // --- async global -> LDS staging: 988 x 16B chunks, direct to LDS ---
  {
    const unsigned ldsBase = (unsigned)(uintptr_t)(void*)&svals[0]; // low 32 bits = LDS address
    for (int i = 0; i < 4; ++i) {
      const int c = tid + i * 256;
      if (c < ASYNC_CHUNKS) {
        const unsigned ldsOff = ldsBase + (unsigned)c * 16u;
        const unsigned long long gaddr = (unsigned long long)(uintptr_t)src
                                       + (unsigned long long)c * 16ull;
        asm volatile("global_load_async_to_lds_b128 %0, %1, off"
                     :: "v"(ldsOff), "v"(gaddr) : "memory");
      }
    }
    // remainder floats (3952..3958) via plain loads
    if (tid < ASYNC_REM) {
      svals[ASYNC_CHUNKS * 4 + tid] = src[ASYNC_CHUNKS * 4 + tid];
    }
    // drain this wave's async copies, then make LDS visible block-wide
    asm volatile("s_wait_asynccnt 0x0" ::: "memory");
  }
  __syncthreads();

  for (int k = 0; k < KTOP; ++k) {
    // per-thread local argmax over strided slice
    float best = -__builtin_huge_valf();
    int bidx = POS;
    for (int i = tid; i < POS; i += 256) {
      const float v = svals[i];
      if (v > best || (v == best && i < bidx)) { best = v; bidx = i; }
    }
    rval[tid] = best;
    ridx[tid] = bidx;
    __syncthreads();

    // tree reduction (value desc, index asc on ties)
    for (int s = 128; s > 0; s >>= 1) {
      if (tid < s) {
        const float ov = rval[tid + s];
        const int   oi = ridx[tid + s];
        if (ov > rval[tid] || (ov == rval[tid] && oi < ridx[tid])) {
          rval[tid] = ov;
          ridx[tid] = oi;
        }
      }
      __syncthreads();
    }

    if (tid == 0) {
      const int   wi = ridx[0];
      const float wv = rval[0];
      svals[wi] = -__builtin_huge_valf();    // knock out winner for next round

      const int elev = wi / A_DIM;
      const int azim = wi % A_DIM;
      const long long o = ((long long)row * KTOP + k) * 4;
      features[o + 0] = (float)azim;         // azimuth
      features[o + 1] = (float)r;            // range
      features[o + 2] = (float)elev;         // elevation
      features[o + 3] = wv;                  // power (= top-k value)
      coords[o + 0] = b;
      coords[o + 1] = elev + 1;
      coords[o + 2] = r;
      coords[o + 3] = azim + 74;
    }
    __syncthreads();
  }
}

// ---------------------------------------------------------------------------
extern "C" void kernel_launch(void* const* d_in, const int* in_sizes, int n_in,
                              void* d_out, int out_size, void* d_ws, size_t ws_size,
                              hipStream_t stream) {
  (void)in_sizes; (void)n_in; (void)out_size; (void)ws_size;

  const float* radar = (const float*)d_in[0];

  float* out      = (float*)d_out;
  float* features = out;                                   // ROWS*KTOP*4 floats
  int*   coords   = (int*)(out + (long long)ROWS * KTOP * 4);

  float* cube = (float*)d_ws;                              // ROWS*POS floats (~8.1 MB)

  const int totalWaves = ROWS * TILES_PER_ROW;             // 126976
  const int blocks1 = (totalWaves + 7) / 8;                // 8 waves / 256-thread block

  doppler_mean_wmma<<<blocks1, 256, 0, stream>>>(radar, cube);
  topk_rows<<<ROWS, 256, 0, stream>>>(cube, features, coords);
}